// LATTICE_54485955117089
// MI455X (gfx1250) — compile-verified
//
#include <hip/hip_runtime.h>
#include <hip/hip_bf16.h>
#include <math.h>

#define N_USERS 20000
#define N_ITEMS 6000
#define EMB     128
#define IMG_DIM 4096
#define TXT_DIM 384
#define TOPK    10
#define N_EDGES 2000000
#define N_NODES (N_USERS + N_ITEMS)
#define LAMBDA  0.9f

typedef __attribute__((ext_vector_type(2))) float v2f;
typedef __attribute__((ext_vector_type(8))) float v8f;

// ---- CDNA5 fp32 WMMA: D(16x16) = A(16x4) * B(4x16) + C ----
__device__ __forceinline__ v8f wmma4(v2f a, v2f b, v8f c) {
    return __builtin_amdgcn_wmma_f32_16x16x4_f32(
        /*neg_a=*/false, a, /*neg_b=*/false, b,
        /*c_mod=*/(short)0, c, /*reuse_a=*/false, /*reuse_b=*/false);
}

__device__ __forceinline__ void lds_fence_wave() {
    asm volatile("s_wait_dscnt 0x0" ::: "memory");
}

__device__ __forceinline__ void softmax2(const float* mw, float& w0, float& w1) {
    float m = fmaxf(mw[0], mw[1]);
    float e0 = expf(mw[0] - m), e1 = expf(mw[1] - m);
    float inv = 1.0f / (e0 + e1);
    w0 = e0 * inv; w1 = e1 * inv;
}

// ============================================================================
// 1) feats = A[M,K] @ W[128,K]^T + bias   (one wave -> 16 rows x 128 cols)
// ============================================================================
__global__ __launch_bounds__(256) void feats_gemm(const float* __restrict__ A,
                                                  const float* __restrict__ W,
                                                  const float* __restrict__ bias,
                                                  float* __restrict__ out,
                                                  int M, int K) {
    const int tid = threadIdx.x, wave = tid >> 5, lane = tid & 31;
    const int half = lane >> 4, l16 = lane & 15;
    const int mt = blockIdx.x * 8 + wave;
    if (mt * 16 >= M) return;
    const int m0 = mt * 16;

    v8f acc[8];
#pragma unroll
    for (int t = 0; t < 8; ++t) acc[t] = (v8f){0,0,0,0,0,0,0,0};

    const size_t arow = (size_t)(m0 + l16) * K;
    for (int k0 = 0; k0 < K; k0 += 4) {
        v2f a = *(const v2f*)(A + arow + k0 + 2 * half);
#pragma unroll
        for (int t = 0; t < 8; ++t) {
            v2f b = *(const v2f*)(W + (size_t)(t * 16 + l16) * K + k0 + 2 * half);
            acc[t] = wmma4(a, b, acc[t]);
        }
    }
#pragma unroll
    for (int t = 0; t < 8; ++t) {
        const int n = t * 16 + l16;
        const float bn = bias[n];
#pragma unroll
        for (int r = 0; r < 8; ++r)
            out[(size_t)(m0 + r + 8 * half) * EMB + n] = acc[t][r] + bn;
    }
}

// ============================================================================
// 2) in-place row L2 normalization (one wave per row of [M,128])
// ============================================================================
__global__ __launch_bounds__(256) void rownorm(float* __restrict__ x, int M) {
    const int wid = (blockIdx.x * blockDim.x + threadIdx.x) >> 5;
    const int lane = threadIdx.x & 31;
    if (wid >= M) return;
    float4 v = *(const float4*)(x + (size_t)wid * EMB + lane * 4);
    float ss = v.x * v.x + v.y * v.y + v.z * v.z + v.w * v.w;
#pragma unroll
    for (int off = 16; off; off >>= 1) ss += __shfl_xor(ss, off, 32);
    float inv = 1.0f / sqrtf(ss);
    v.x *= inv; v.y *= inv; v.z *= inv; v.w *= inv;
    *(float4*)(x + (size_t)wid * EMB + lane * 4) = v;
}

// ============================================================================
// 3) fused sim + top-10: block = 16-row strip; waves sweep column tiles with
//    WMMA, per-wave top-10 lists in LDS, merged by lanes of wave 0 at the end.
//    Never materializes the 6000x6000 sim matrix.
// ============================================================================
__global__ __launch_bounds__(256) void sim_topk(const float* __restrict__ xn,
                                                float* __restrict__ kvals,
                                                int* __restrict__ kcols) {
    __shared__ float rows_s[16][EMB];        // 8 KB: row strip, WMMA A source
    __shared__ float tile[8][16][16];        // 8 KB: per-wave 16x16 staging
    __shared__ float twv[8][16][TOPK];       // 5 KB
    __shared__ int   twc[8][16][TOPK];       // 5 KB
    const int tid = threadIdx.x, wave = tid >> 5, lane = tid & 31;
    const int half = lane >> 4, l16 = lane & 15;
    const int m0 = blockIdx.x * 16;

    for (int i = tid; i < 16 * EMB; i += 256)
        rows_s[i >> 7][i & 127] = xn[(size_t)(m0 + (i >> 7)) * EMB + (i & 127)];
    if (lane < 16)
#pragma unroll
        for (int j = 0; j < TOPK; ++j) { twv[wave][lane][j] = -__builtin_inff(); twc[wave][lane][j] = 0; }
    __syncthreads();

    const int NT = N_ITEMS / 16;             // 375 column tiles
    for (int it = 0; it < (NT + 7) / 8; ++it) {
        const int nt = it * 8 + wave;
        const bool valid = nt < NT;
        const int n0 = valid ? nt * 16 : 0;  // clamp keeps loads in-range
        v8f acc = (v8f){0,0,0,0,0,0,0,0};
        for (int k0 = 0; k0 < EMB; k0 += 4) {
            v2f a = *(const v2f*)(&rows_s[l16][k0 + 2 * half]);
            v2f b = *(const v2f*)(xn + (size_t)(n0 + l16) * EMB + k0 + 2 * half);
            acc = wmma4(a, b, acc);
        }
#pragma unroll
        for (int r = 0; r < 8; ++r) tile[wave][r + 8 * half][l16] = acc[r];
        lds_fence_wave();                    // order tile stores vs. cross-lane reads
        if (valid && lane < 16) {
            const int r = lane;
            float* tv = twv[wave][r]; int* tc = twc[wave][r];
            for (int c = 0; c < 16; ++c) {
                const float v = tile[wave][r][c];
                if (v > tv[TOPK - 1]) {
                    int p = TOPK - 1;
                    while (p > 0 && tv[p - 1] < v) { tv[p] = tv[p - 1]; tc[p] = tc[p - 1]; --p; }
                    tv[p] = v; tc[p] = n0 + c;
                }
            }
        }
    }
    __syncthreads();

    if (tid < 16) {                          // merge the 8 wave-local lists
        const int r = tid;
        float bv[TOPK]; int bc[TOPK];
#pragma unroll
        for (int j = 0; j < TOPK; ++j) { bv[j] = -__builtin_inff(); bc[j] = 0; }
        for (int w = 0; w < 8; ++w)
            for (int j = 0; j < TOPK; ++j) {
                const float v = twv[w][r][j]; const int c = twc[w][r][j];
                if (v > bv[TOPK - 1]) {
                    int p = TOPK - 1;
                    while (p > 0 && bv[p - 1] < v) { bv[p] = bv[p - 1]; bc[p] = bc[p - 1]; --p; }
                    bv[p] = v; bc[p] = c;
                }
            }
        for (int j = 0; j < TOPK; ++j) {
            kvals[(size_t)(m0 + r) * TOPK + j] = bv[j];
            kcols[(size_t)(m0 + r) * TOPK + j] = bc[j];
        }
    }
}

// ============================================================================
// 4) combine modal kNN lists (w-scaled), compute rowsum -> d^{-1/2}
// ============================================================================
__global__ __launch_bounds__(256) void combine_knn(const float* __restrict__ kvi,
                                                   const int* __restrict__ kci,
                                                   const float* __restrict__ kvt,
                                                   const int* __restrict__ kct,
                                                   const float* __restrict__ mw,
                                                   float* __restrict__ cvals,
                                                   int* __restrict__ ccols,
                                                   float* __restrict__ dinv) {
    const int r = blockIdx.x * blockDim.x + threadIdx.x;
    if (r >= N_ITEMS) return;
    float w0, w1; softmax2(mw, w0, w1);
    float rs = 0.0f;
#pragma unroll
    for (int j = 0; j < TOPK; ++j) {
        const float v = w0 * kvi[(size_t)r * TOPK + j];
        cvals[(size_t)r * 2 * TOPK + j] = v;
        ccols[(size_t)r * 2 * TOPK + j] = kci[(size_t)r * TOPK + j];
        rs += v;
    }
#pragma unroll
    for (int j = 0; j < TOPK; ++j) {
        const float v = w1 * kvt[(size_t)r * TOPK + j];
        cvals[(size_t)r * 2 * TOPK + TOPK + j] = v;
        ccols[(size_t)r * 2 * TOPK + TOPK + j] = kct[(size_t)r * TOPK + j];
        rs += v;
    }
    dinv[r] = (rs > 0.0f) ? rsqrtf(rs) : 0.0f;
}

// ============================================================================
// 5a) h = LAMBDA * (w0*IOA + w1*TOA) @ item_emb   (dense WMMA GEMM,
//     streams both 144MB matrices exactly once — HBM-bound, ~12us)
// ============================================================================
__global__ __launch_bounds__(256) void dense_item_gemm(const float* __restrict__ ioa,
                                                       const float* __restrict__ toa,
                                                       const float* __restrict__ ie,
                                                       const float* __restrict__ mw,
                                                       float* __restrict__ h) {
    const int tid = threadIdx.x, wave = tid >> 5, lane = tid & 31;
    const int half = lane >> 4, l16 = lane & 15;
    const int mt = blockIdx.x * 8 + wave;
    if (mt * 16 >= N_ITEMS) return;
    float w0, w1; softmax2(mw, w0, w1);
    const float s0 = LAMBDA * w0, s1 = LAMBDA * w1;
    const int m0 = mt * 16;

    v8f acc[8];
#pragma unroll
    for (int t = 0; t < 8; ++t) acc[t] = (v8f){0,0,0,0,0,0,0,0};

    const size_t rowb = (size_t)(m0 + l16) * N_ITEMS;
    for (int k0 = 0; k0 < N_ITEMS; k0 += 4) {
        const size_t ai = rowb + k0 + 2 * half;
        __builtin_prefetch(ioa + ai + 256, 0, 1);
        __builtin_prefetch(toa + ai + 256, 0, 1);
        v2f ia = *(const v2f*)(ioa + ai);
        v2f ta = *(const v2f*)(toa + ai);
        v2f a; a.x = s0 * ia.x + s1 * ta.x; a.y = s0 * ia.y + s1 * ta.y;
        const int kr = k0 + 2 * half;
#pragma unroll
        for (int t = 0; t < 8; ++t) {
            v2f b;
            b.x = ie[(size_t)kr * EMB + t * 16 + l16];
            b.y = ie[(size_t)(kr + 1) * EMB + t * 16 + l16];
            acc[t] = wmma4(a, b, acc[t]);
        }
    }
#pragma unroll
    for (int t = 0; t < 8; ++t)
#pragma unroll
        for (int r = 0; r < 8; ++r)
            h[(size_t)(m0 + r + 8 * half) * EMB + t * 16 + l16] = acc[t][r];
}

// ============================================================================
// 5b) h += (1-LAMBDA) * D^{-1/2} A_learned D^{-1/2} @ item_emb  (<=20 nz/row)
// ============================================================================
__global__ __launch_bounds__(256) void learned_sparse_add(const float* __restrict__ cvals,
                                                          const int* __restrict__ ccols,
                                                          const float* __restrict__ dinv,
                                                          const float* __restrict__ ie,
                                                          float* __restrict__ h) {
    const int wid = (blockIdx.x * blockDim.x + threadIdx.x) >> 5;
    const int lane = threadIdx.x & 31;
    if (wid >= N_ITEMS) return;
    const float di = (1.0f - LAMBDA) * dinv[wid];
    float4 acc = {0, 0, 0, 0};
#pragma unroll
    for (int e = 0; e < 2 * TOPK; ++e) {
        const float v = cvals[(size_t)wid * 2 * TOPK + e];
        const int   c = ccols[(size_t)wid * 2 * TOPK + e];
        const float s = di * v * dinv[c];
        const float4 g = *(const float4*)(ie + (size_t)c * EMB + lane * 4);
        acc.x += s * g.x; acc.y += s * g.y; acc.z += s * g.z; acc.w += s * g.w;
    }
    float4* hp = (float4*)(h + (size_t)wid * EMB + lane * 4);
    float4 o = *hp;
    o.x += acc.x; o.y += acc.y; o.z += acc.z; o.w += acc.w;
    *hp = o;
}

// ============================================================================
// 6) ego0 = concat(user_emb, item_emb); acc = ego0
// ============================================================================
__global__ __launch_bounds__(256) void init_ego(const float* __restrict__ ue,
                                                const float* __restrict__ ie,
                                                float* __restrict__ ego,
                                                float* __restrict__ accb) {
    const size_t i = (size_t)blockIdx.x * blockDim.x + threadIdx.x;
    if (i >= (size_t)N_NODES * EMB) return;
    const float v = (i < (size_t)N_USERS * EMB) ? ue[i] : ie[i - (size_t)N_USERS * EMB];
    ego[i] = v; accb[i] = v;
}

__global__ __launch_bounds__(256) void zero_buf(float* __restrict__ p, size_t n) {
    const size_t i = (size_t)blockIdx.x * blockDim.x + threadIdx.x;
    if (i < n) p[i] = 0.0f;
}

// ============================================================================
// 7) one SpMM layer: out[row] += val * ego[col]  (wave per edge, f32 atomics)
// ============================================================================
__global__ __launch_bounds__(256) void edge_spmm(const float* __restrict__ vals,
                                                 const int* __restrict__ rows,
                                                 const int* __restrict__ cols,
                                                 const float* __restrict__ ego,
                                                 float* __restrict__ outp) {
    const size_t gid = (size_t)blockIdx.x * blockDim.x + threadIdx.x;
    const size_t e = gid >> 5;
    const int lane = threadIdx.x & 31;
    if (e >= (size_t)N_EDGES) return;
    const float v = vals[e];
    const int r = rows[e], c = cols[e];
    const float4 g = *(const float4*)(ego + (size_t)c * EMB + lane * 4);
    float* o = outp + (size_t)r * EMB + lane * 4;
    unsafeAtomicAdd(o + 0, v * g.x);
    unsafeAtomicAdd(o + 1, v * g.y);
    unsafeAtomicAdd(o + 2, v * g.z);
    unsafeAtomicAdd(o + 3, v * g.w);
}

__global__ __launch_bounds__(256) void acc_add(float* __restrict__ accb,
                                               const float* __restrict__ x, size_t n) {
    const size_t i = (size_t)blockIdx.x * blockDim.x + threadIdx.x;
    if (i < n) accb[i] += x[i];
}

// ============================================================================
// 8) finalize: u_g = acc/3 ; i_g = acc/3 + h/max(||h||,eps)
// ============================================================================
__global__ __launch_bounds__(256) void final_users(const float* __restrict__ accb,
                                                   float* __restrict__ out) {
    const size_t i = (size_t)blockIdx.x * blockDim.x + threadIdx.x;
    if (i < (size_t)N_USERS * EMB) out[i] = accb[i] * (1.0f / 3.0f);
}

__global__ __launch_bounds__(256) void final_items(const float* __restrict__ accb,
                                                   const float* __restrict__ h,
                                                   float* __restrict__ out) {
    const int wid = (blockIdx.x * blockDim.x + threadIdx.x) >> 5;
    const int lane = threadIdx.x & 31;
    if (wid >= N_ITEMS) return;
    const float4 hv = *(const float4*)(h + (size_t)wid * EMB + lane * 4);
    float ss = hv.x * hv.x + hv.y * hv.y + hv.z * hv.z + hv.w * hv.w;
#pragma unroll
    for (int off = 16; off; off >>= 1) ss += __shfl_xor(ss, off, 32);
    const float inv = 1.0f / fmaxf(sqrtf(ss), 1e-12f);
    const size_t base = (size_t)(N_USERS + wid) * EMB + lane * 4;
    float4 a = *(const float4*)(accb + base);
    float4 o;
    o.x = a.x * (1.0f / 3.0f) + hv.x * inv;
    o.y = a.y * (1.0f / 3.0f) + hv.y * inv;
    o.z = a.z * (1.0f / 3.0f) + hv.z * inv;
    o.w = a.w * (1.0f / 3.0f) + hv.w * inv;
    *(float4*)(out + base) = o;
}

// ============================================================================
extern "C" void kernel_launch(void* const* d_in, const int* in_sizes, int n_in,
                              void* d_out, int out_size, void* d_ws, size_t ws_size,
                              hipStream_t stream) {
    const float* user_emb = (const float*)d_in[0];
    const float* item_emb = (const float*)d_in[1];
    const float* image_emb = (const float*)d_in[2];
    const float* text_emb = (const float*)d_in[3];
    const float* W_img = (const float*)d_in[4];
    const float* b_img = (const float*)d_in[5];
    const float* W_txt = (const float*)d_in[6];
    const float* b_txt = (const float*)d_in[7];
    const float* modal_w = (const float*)d_in[8];
    const float* ioa = (const float*)d_in[9];
    const float* toa = (const float*)d_in[10];
    const float* adj_vals = (const float*)d_in[11];
    const int* adj_rows = (const int*)d_in[12];
    const int* adj_cols = (const int*)d_in[13];
    float* out = (float*)d_out;

    // ---- workspace layout (floats), ~51 MB total ----
    float* w = (float*)d_ws;
    float* fimg  = w;                          // 6000*128
    float* ftxt  = fimg + 768000;              // 6000*128
    float* kvi   = ftxt + 768000;              // 6000*10
    float* kvt   = kvi + 60000;
    float* cvals = kvt + 60000;                // 6000*20
    float* dinv  = cvals + 120000;             // 6000
    float* h     = dinv + 6000;                // 6000*128
    float* ego_a = h + 768000;                 // 26000*128
    float* ego_b = ego_a + 3328000;
    float* accb  = ego_b + 3328000;
    int* ibase = (int*)(accb + 3328000);
    int* kci   = ibase;                        // 6000*10
    int* kct   = kci + 60000;
    int* ccols = kct + 60000;                  // 6000*20

    const size_t nemb = (size_t)N_NODES * EMB; // 3,328,000
    const int EW_BLK = (int)((nemb + 255) / 256);

    // 1) modal feature projections (WMMA f32)
    feats_gemm<<<47, 256, 0, stream>>>(image_emb, W_img, b_img, fimg, N_ITEMS, IMG_DIM);
    feats_gemm<<<47, 256, 0, stream>>>(text_emb, W_txt, b_txt, ftxt, N_ITEMS, TXT_DIM);

    // 2) row-normalize for cosine sim
    rownorm<<<750, 256, 0, stream>>>(fimg, N_ITEMS);
    rownorm<<<750, 256, 0, stream>>>(ftxt, N_ITEMS);

    // 3) fused sim GEMM + top-10 (WMMA), per modality
    sim_topk<<<375, 256, 0, stream>>>(fimg, kvi, kci);
    sim_topk<<<375, 256, 0, stream>>>(ftxt, kvt, kct);

    // 4) combine + normalized-laplacian degrees
    combine_knn<<<24, 256, 0, stream>>>(kvi, kci, kvt, kct, modal_w, cvals, ccols, dinv);

    // 5) h = item_adj @ item_emb  (dense WMMA part + sparse learned part)
    dense_item_gemm<<<47, 256, 0, stream>>>(ioa, toa, item_emb, modal_w, h);
    learned_sparse_add<<<750, 256, 0, stream>>>(cvals, ccols, dinv, item_emb, h);

    // 6) ego0 = concat(users, items); acc = ego0
    init_ego<<<EW_BLK, 256, 0, stream>>>(user_emb, item_emb, ego_a, accb);

    // 7) two UI propagation layers (segment_sum via f32 atomics)
    const int EDGE_BLK = (int)(((size_t)N_EDGES * 32 + 255) / 256);
    zero_buf<<<EW_BLK, 256, 0, stream>>>(ego_b, nemb);
    edge_spmm<<<EDGE_BLK, 256, 0, stream>>>(adj_vals, adj_rows, adj_cols, ego_a, ego_b);
    acc_add<<<EW_BLK, 256, 0, stream>>>(accb, ego_b, nemb);
    zero_buf<<<EW_BLK, 256, 0, stream>>>(ego_a, nemb);
    edge_spmm<<<EDGE_BLK, 256, 0, stream>>>(adj_vals, adj_rows, adj_cols, ego_b, ego_a);
    acc_add<<<EW_BLK, 256, 0, stream>>>(accb, ego_a, nemb);

    // 8) outputs
    final_users<<<10000, 256, 0, stream>>>(accb, out);
    final_items<<<750, 256, 0, stream>>>(accb, h, out);
}